// MultiLoRALinear_65687229825758
// MI455X (gfx1250) — compile-verified
//
#include <hip/hip_runtime.h>

// ---------------------------------------------------------------------------
// LoRA linear for MI455X (gfx1250, wave32):
//   out[m,n] = sum_k x[m,k]*W[n,k] + SCALE * sum_r (x@A^T)[m,r]*loraB[n,r]
// M=8192, N=4096, K=4096, R=8, fp32.
// Core: V_WMMA_F32_16X16X4_F32 bursts + GLOBAL_LOAD_ASYNC_TO_LDS_B128
// double-buffered staging (ASYNCcnt), nontemporal output stores.
// ---------------------------------------------------------------------------

typedef __attribute__((ext_vector_type(2))) float v2f;
typedef __attribute__((ext_vector_type(8))) float v8f;
typedef __attribute__((address_space(3))) float lds_float_t;

#define LORA_SCALE 2.0f   // alpha/rank = 16/8

constexpr int D_IN  = 4096;
constexpr int D_OUT = 4096;
constexpr int R     = 8;

constexpr int BM  = 128;    // block tile M
constexpr int BN  = 128;    // block tile N
constexpr int BK  = 32;     // block tile K
constexpr int LDK = BK + 4; // padded LDS row: 36 floats = 144B (16B-aligned rows,
                            // stride 36 conflict-free across 64 banks)
constexpr int TILE_F = BM * LDK;  // floats per (matrix, buffer) tile

// Async copy 16B global -> LDS (per lane), tracked by ASYNCcnt.
__device__ __forceinline__ void async_b128(lds_float_t* lds, const float* g) {
    asm volatile("global_load_async_to_lds_b128 %0, %1, off"
                 :: "v"(lds), "v"(g) : "memory");
}

// ---------------------------------------------------------------------------
// Kernel 1: xa[m][r] = SCALE * dot(x[m,:], lora_A[r,:]);  one wave per row.
// ---------------------------------------------------------------------------
__global__ void __launch_bounds__(256)
lora_xa_kernel(const float* __restrict__ x,
               const float* __restrict__ lora_A,
               float* __restrict__ xa, int M) {
    const int wave = (int)((blockIdx.x * blockDim.x + threadIdx.x) >> 5);
    const int lane = (int)(threadIdx.x & 31);
    if (wave >= M) return;

    const float* xrow = x + (size_t)wave * D_IN;
    float acc[R];
#pragma unroll
    for (int r = 0; r < R; ++r) acc[r] = 0.0f;

    for (int k = lane; k < D_IN; k += 32) {
        const float xv = xrow[k];
#pragma unroll
        for (int r = 0; r < R; ++r)
            acc[r] += xv * lora_A[r * D_IN + k];
    }
#pragma unroll
    for (int r = 0; r < R; ++r) {
        float v = acc[r];
#pragma unroll
        for (int off = 16; off > 0; off >>= 1)
            v += __shfl_xor(v, off, 32);
        if (lane == 0) xa[(size_t)wave * R + r] = v * LORA_SCALE;
    }
}

// ---------------------------------------------------------------------------
// Kernel 2: fused GEMM.  8 waves/block (2 M x 4 N), each wave owns 64x32 =
// 4x2 fp32 16x16 fragments.  Double-buffered async LDS staging.
// ---------------------------------------------------------------------------
__global__ void __launch_bounds__(256)
lora_gemm_kernel(const float* __restrict__ x,      // [M,  D_IN ]
                 const float* __restrict__ w,      // [N,  D_IN ]
                 const float* __restrict__ xa,     // [M,  R] (pre-scaled)
                 const float* __restrict__ loraB,  // [N,  R]
                 float* __restrict__ out) {        // [M,  N]
    __shared__ float lx[2 * TILE_F];
    __shared__ float lw[2 * TILE_F];

    const int tid  = (int)threadIdx.x;
    const int lane = tid & 31;
    const int wave = tid >> 5;     // 0..7
    const int wm   = wave & 1;     // wave M coord (64 rows)
    const int wn   = wave >> 1;    // wave N coord (32 cols)

    const int blockM = (int)blockIdx.y * BM;
    const int blockN = (int)blockIdx.x * BN;

    // Fragment lane coords (ISA 16x4 fp32 A/B layout):
    const int fr = lane & 15;            // row (A) / col (B) within 16
    const int kh = (lane >> 4) << 1;     // k pair base: 0 or 2

    // Async staging assignment: chunk c = tid + i*256 (i=0..3) per matrix;
    // row = c>>3 (0..127), col = (c&7)*4.  16B-aligned LDS targets.
    const int srow = tid >> 3;           // base row (0..31)
    const int scol = (tid & 7) << 2;     // 0,4,...,28
    const float* gx = x + (size_t)(blockM + srow) * D_IN + scol;
    const float* gw = w + (size_t)(blockN + srow) * D_IN + scol;

    v8f acc[4][2] = {};

    // Prologue: stage tile k0=0 into buffer 0.
#pragma unroll
    for (int i = 0; i < 4; ++i) {
        const int l = (srow + i * 32) * LDK + scol;
        async_b128((lds_float_t*)&lx[l], gx + (size_t)i * 32 * D_IN);
        async_b128((lds_float_t*)&lw[l], gw + (size_t)i * 32 * D_IN);
    }

    int buf = 0;
    for (int k0 = 0; k0 < D_IN; k0 += BK) {
        // Stage next tile into the other buffer while this one computes.
        if (k0 + BK < D_IN) {
            const int nb = (buf ^ 1) * TILE_F;
#pragma unroll
            for (int i = 0; i < 4; ++i) {
                const int l = nb + (srow + i * 32) * LDK + scol;
                async_b128((lds_float_t*)&lx[l], gx + (size_t)i * 32 * D_IN + (k0 + BK));
                async_b128((lds_float_t*)&lw[l], gw + (size_t)i * 32 * D_IN + (k0 + BK));
            }
            // 8 just issued may remain; the previous 8 (this tile) must be done.
            asm volatile("s_wait_asynccnt 0x8" ::: "memory");
        } else {
            asm volatile("s_wait_asynccnt 0x0" ::: "memory");
        }
        __syncthreads();   // this tile visible to all waves

        const float* lxb = &lx[buf * TILE_F];
        const float* lwb = &lw[buf * TILE_F];

        // Two bursts of 4 k-steps: preload all fragments, then 32 WMMAs.
#pragma unroll
        for (int h = 0; h < 2; ++h) {
            v2f a[4][4], b[4][2];
#pragma unroll
            for (int q = 0; q < 4; ++q) {
                const int ks = h * 16 + q * 4 + kh;
#pragma unroll
                for (int i = 0; i < 4; ++i)
                    a[q][i] = *(const v2f*)&lxb[(wm * 64 + i * 16 + fr) * LDK + ks];
#pragma unroll
                for (int j = 0; j < 2; ++j)
                    b[q][j] = *(const v2f*)&lwb[(wn * 32 + j * 16 + fr) * LDK + ks];
            }
#pragma unroll
            for (int q = 0; q < 4; ++q)
#pragma unroll
                for (int i = 0; i < 4; ++i)
#pragma unroll
                    for (int j = 0; j < 2; ++j)
                        acc[i][j] = __builtin_amdgcn_wmma_f32_16x16x4_f32(
                            false, a[q][i], false, b[q][j],
                            (short)0, acc[i][j], false, false);
        }
        __syncthreads();   // all reads of this buffer done before it is refilled
        buf ^= 1;
    }

    // LoRA tail: K'=8 from pre-scaled xa and loraB (tiny, L2-resident).
#pragma unroll
    for (int ks = 0; ks < R; ks += 4) {
        v2f a[4], b[2];
#pragma unroll
        for (int i = 0; i < 4; ++i) {
            const int m = blockM + wm * 64 + i * 16 + fr;
            a[i] = *(const v2f*)&xa[(size_t)m * R + ks + kh];
        }
#pragma unroll
        for (int j = 0; j < 2; ++j) {
            const int n = blockN + wn * 32 + j * 16 + fr;
            b[j] = *(const v2f*)&loraB[(size_t)n * R + ks + kh];
        }
#pragma unroll
        for (int i = 0; i < 4; ++i)
#pragma unroll
            for (int j = 0; j < 2; ++j)
                acc[i][j] = __builtin_amdgcn_wmma_f32_16x16x4_f32(
                    false, a[i], false, b[j], (short)0, acc[i][j], false, false);
    }

    // Store (nontemporal: don't evict L2-resident weights).
    // C/D layout: VGPR v holds M = v + 8*(lane>>4), N = lane&15.
    const int half = lane >> 4;
#pragma unroll
    for (int i = 0; i < 4; ++i) {
        const int mBase = blockM + wm * 64 + i * 16 + half * 8;
#pragma unroll
        for (int j = 0; j < 2; ++j) {
            const int n = blockN + wn * 32 + j * 16 + fr;
#pragma unroll
            for (int v = 0; v < 8; ++v)
                __builtin_nontemporal_store(acc[i][j][v],
                                            &out[(size_t)(mBase + v) * D_OUT + n]);
        }
    }
}

// ---------------------------------------------------------------------------
extern "C" void kernel_launch(void* const* d_in, const int* in_sizes, int n_in,
                              void* d_out, int out_size, void* d_ws, size_t ws_size,
                              hipStream_t stream) {
    const float* x      = (const float*)d_in[0];   // [B,S,D_IN]
    const float* weight = (const float*)d_in[1];   // [D_OUT,D_IN]
    const float* lora_A = (const float*)d_in[2];   // [R,D_IN]
    const float* lora_B = (const float*)d_in[3];   // [D_OUT,R]
    float*       out    = (float*)d_out;           // [B,S,D_OUT]
    float*       xa     = (float*)d_ws;            // [M,R] scratch (256 KB)

    const int M = in_sizes[0] / D_IN;              // 8192

    lora_xa_kernel<<<dim3((M + 7) / 8), 256, 0, stream>>>(x, lora_A, xa, M);

    dim3 grid(D_OUT / BN, M / BM);
    lora_gemm_kernel<<<grid, 256, 0, stream>>>(x, weight, xa, lora_B, out);
}